// Select_Profile_37958920962575
// MI455X (gfx1250) — compile-verified
//
#include <hip/hip_runtime.h>
#include <cstdint>

#define N_SAMP 131072
#define S_PSF  64
#define DIMV   256           // D = H = W = 256
#define WAVES_PER_BLOCK 8
#define BLOCK  (WAVES_PER_BLOCK * 32)
#define STAGE_FLOATS (WAVES_PER_BLOCK * S_PSF * 3)   // 1536 floats = 6144 B

// ---- gfx1250 async global->LDS path (guarded; falls back to float4 staging) ----
#ifndef USE_ASYNC_LDS
#define USE_ASYNC_LDS 0
#endif
#if defined(__has_builtin)
#  if __has_builtin(__builtin_amdgcn_global_load_async_to_lds_b128)
#    undef  USE_ASYNC_LDS
#    define USE_ASYNC_LDS 1
#  endif
#endif

// cpol immediate: TH[2:0] | scope[4:3]; TH_LOAD_NT = 1 (stream xyz_psf past L2,
// keep the 64MB volume resident in the 192MB L2)
#define ASYNC_CPOL 1

typedef int v4i __attribute__((__vector_size__(4 * sizeof(int))));
typedef __attribute__((address_space(1))) v4i gas_v4i;
typedef __attribute__((address_space(3))) v4i las_v4i;

__device__ __forceinline__ float wave_reduce_add(float v) {
#pragma unroll
    for (int off = 16; off > 0; off >>= 1)
        v += __shfl_xor(v, off, 32);
    return v;
}

__global__ __launch_bounds__(BLOCK)
void psf_sample_kernel(const float* __restrict__ vol,        // 256^3
                       const float* __restrict__ sampleGrid, // N,3
                       const float* __restrict__ ax,         // N,6 (rot|trn)
                       const float* __restrict__ bound,      // N,2,3
                       const float* __restrict__ invCov,     // 3,3
                       const float* __restrict__ xyzPsf,     // N,S,3
                       float* __restrict__ out)              // N
{
    __shared__ float sXyz[STAGE_FLOATS];   // 6144 B: xyz_psf for this block's 8 samples

    const int tid    = threadIdx.x;
    const int lane   = tid & 31;
    const int waveId = tid >> 5;
    const int n0     = blockIdx.x * WAVES_PER_BLOCK;
    const int n      = n0 + waveId;

    // ---- stage xyz_psf for 8 samples into LDS (coalesced b128, non-temporal) ----
    const float* src = xyzPsf + (size_t)n0 * (S_PSF * 3);
#if USE_ASYNC_LDS
    // 6144 B = 384 x b128 transfers; ASYNCcnt-tracked, bypasses VGPRs.
#pragma unroll 2
    for (int i = tid; i < STAGE_FLOATS / 4; i += BLOCK) {
        __builtin_amdgcn_global_load_async_to_lds_b128(
            (gas_v4i*)(uintptr_t)(src + i * 4),
            (las_v4i*)(uintptr_t)(&sXyz[i * 4]),
            0, ASYNC_CPOL);
    }
#  if __has_builtin(__builtin_amdgcn_s_wait_asynccnt)
    __builtin_amdgcn_s_wait_asynccnt(0);
#  else
    asm volatile("s_wait_asynccnt 0x0" ::: "memory");
#  endif
    __syncthreads();
#else
#pragma unroll 2
    for (int i = tid * 4; i < STAGE_FLOATS; i += BLOCK * 4) {
        float4 v = *(const float4*)(src + i);
        *(float4*)(&sXyz[i]) = v;
    }
    __syncthreads();
#endif

    // ---- per-sample setup (wave-uniform -> scalar loads/ALU) ----
    const float rx = ax[n * 6 + 0], ry = ax[n * 6 + 1], rz = ax[n * 6 + 2];
    const float tx = ax[n * 6 + 3], ty = ax[n * 6 + 4], tz = ax[n * 6 + 5];

    float theta = sqrtf(rx * rx + ry * ry + rz * rz);
    float kinv  = 1.0f / fmaxf(theta, 1e-12f);
    float kx = rx * kinv, ky = ry * kinv, kz = rz * kinv;
    float st, ct;
    __sincosf(theta, &st, &ct);
    float omc = 1.0f - ct;

    // R = I + st*K + (1-ct)*K^2, K = [[0,-kz,ky],[kz,0,-kx],[-ky,kx,0]]
    float kxx = kx * kx, kyy = ky * ky, kzz = kz * kz;
    float kxy = kx * ky, kxz = kx * kz, kyz = ky * kz;
    float r00 = 1.0f - omc * (kyy + kzz);
    float r01 = -st * kz + omc * kxy;
    float r02 =  st * ky + omc * kxz;
    float r10 =  st * kz + omc * kxy;
    float r11 = 1.0f - omc * (kxx + kzz);
    float r12 = -st * kx + omc * kyz;
    float r20 = -st * ky + omc * kxz;
    float r21 =  st * kx + omc * kyz;
    float r22 = 1.0f - omc * (kxx + kyy);

    float px = sampleGrid[n * 3 + 0] + tx;
    float py = sampleGrid[n * 3 + 1] + ty;
    float pz = sampleGrid[n * 3 + 2] + tz;
    float mx = r00 * px + r01 * py + r02 * pz;
    float my = r10 * px + r11 * py + r12 * pz;
    float mz = r20 * px + r21 * py + r22 * pz;

    float hx = (bound[n * 6 + 3] - bound[n * 6 + 0]) * 0.5f;
    float hy = (bound[n * 6 + 4] - bound[n * 6 + 1]) * 0.5f;
    float hz = (bound[n * 6 + 5] - bound[n * 6 + 2]) * 0.5f;

    float ic00 = invCov[0], ic01 = invCov[1], ic02 = invCov[2];
    float ic10 = invCov[3], ic11 = invCov[4], ic12 = invCov[5];
    float ic20 = invCov[6], ic21 = invCov[7], ic22 = invCov[8];

    // ---- each lane handles s = lane and s = lane + 32 ----
    float num = 0.0f, den = 0.0f;
    const float* sbase = &sXyz[waveId * (S_PSF * 3)];

#pragma unroll
    for (int half = 0; half < 2; ++half) {
        const int s = lane + half * 32;
        float ux = sbase[s * 3 + 0];
        float uy = sbase[s * 3 + 1];
        float uz = sbase[s * 3 + 2];

        float bx = ux * hx, by = uy * hy, bz = uz * hz;   // bound_psf (== center)
        float fx3 = mx + bx, fy3 = my + by, fz3 = mz + bz; // psf coords

        // grid = 2*(psf/(size-1) - 0.5); pix = ((grid+1)*size - 1)*0.5
        const float invSm1 = 1.0f / 255.0f;
        float gx = 2.0f * (fx3 * invSm1 - 0.5f);
        float gy = 2.0f * (fy3 * invSm1 - 0.5f);
        float gz = 2.0f * (fz3 * invSm1 - 0.5f);
        float ixf = ((gx + 1.0f) * 256.0f - 1.0f) * 0.5f;
        float iyf = ((gy + 1.0f) * 256.0f - 1.0f) * 0.5f;
        float izf = ((gz + 1.0f) * 256.0f - 1.0f) * 0.5f;

        float x0f = floorf(ixf), y0f = floorf(iyf), z0f = floorf(izf);
        float fx = ixf - x0f, fy = iyf - y0f, fz = izf - z0f;
        int x0 = (int)x0f, y0 = (int)y0f, z0 = (int)z0f;

        // x-pair is contiguous: fetch both x-taps of each (z,y) row with ONE
        // global_load_b64 (4B-aligned, legal in DWORD/UNALIGNED modes).
        int   xc  = x0 < 0 ? 0 : (x0 > 254 ? 254 : x0);   // pair base in [0,254]
        bool  vx0 = ((unsigned)x0 < 256u);
        bool  vx1 = ((unsigned)(x0 + 1) < 256u);
        bool  selHi = (x0 == 255);   // value@x0   is pair.y when clamped down
        bool  selLo = (x0 == -1);    // value@x0+1 is pair.x when clamped up
        float wx1 = fx, wx0 = 1.0f - fx;

        float val = 0.0f;
#pragma unroll
        for (int dz = 0; dz < 2; ++dz) {
            int   zi = z0 + dz;
            float wz = dz ? fz : 1.0f - fz;
#pragma unroll
            for (int dy = 0; dy < 2; ++dy) {
                int   yi = y0 + dy;
                float wy = dy ? fy : 1.0f - fy;
                bool  rowValid = ((unsigned)yi < 256u) & ((unsigned)zi < 256u);
                float2 p = make_float2(0.0f, 0.0f);
                if (rowValid)
                    p = *(const float2*)(vol + ((zi << 8) + yi) * 256 + xc);
                float v0 = selHi ? p.y : p.x;   // tap at x0
                float v1 = selLo ? p.x : p.y;   // tap at x0+1
                v0 = vx0 ? v0 : 0.0f;
                v1 = vx1 ? v1 : 0.0f;
                val = fmaf(fmaf(v0, wx0, v1 * wx1), wy * wz, val);
            }
        }

        // q = b^T * IC * b ; w = exp(-0.5 q)
        float q = bx * (ic00 * bx + ic01 * by + ic02 * bz)
                + by * (ic10 * bx + ic11 * by + ic12 * bz)
                + bz * (ic20 * bx + ic21 * by + ic22 * bz);
        float wgt = __expf(-0.5f * q);

        num = fmaf(val, wgt, num);
        den += wgt;
    }

    num = wave_reduce_add(num);
    den = wave_reduce_add(den);
    if (lane == 0) out[n] = num / den;
}

extern "C" void kernel_launch(void* const* d_in, const int* in_sizes, int n_in,
                              void* d_out, int out_size, void* d_ws, size_t ws_size,
                              hipStream_t stream) {
    const float* vol = (const float*)d_in[0];  // x           (1,1,256,256,256)
    const float* sg  = (const float*)d_in[1];  // sampleGrid  (N,3)
    const float* axp = (const float*)d_in[2];  // ax          (N,1,6)
    const float* bnd = (const float*)d_in[3];  // bound       (N,2,3)
    const float* ic  = (const float*)d_in[4];  // InvCovScaled(3,3)
    const float* xp  = (const float*)d_in[5];  // xyz_psf     (N,S,3)
    float* out = (float*)d_out;                // y           (N,)

    dim3 grid(N_SAMP / WAVES_PER_BLOCK);
    psf_sample_kernel<<<grid, BLOCK, 0, stream>>>(vol, sg, axp, bnd, ic, xp, out);
}